// AttentionDecoder_10170482557153
// MI455X (gfx1250) — compile-verified
//
#include <hip/hip_runtime.h>
#include <math.h>

#define Hdim 1024
#define Ldim 512
#define Vdim 50257

typedef __attribute__((ext_vector_type(2))) float v2f;
typedef __attribute__((ext_vector_type(8))) float v8f;

// ---------------------------------------------------------------------------
// K-split WMMA GEMV partial:  handles rows [blockIdx.x*128, +128) over the
// K-window [blockIdx.y*(K/ksplit), +K/ksplit) using V_WMMA_F32_16X16X4_F32.
//   ksplit == 1 : y[row]                = dot + bias[row]   (final store)
//   ksplit  > 1 : y[blockIdx.y*R + row] = partial dot       (no bias)
// Each wave owns a 16-row tile; B operand = x chunk broadcast to all 16
// columns (every column of D holds the same dot; we read column N=0).
// klen must be a multiple of 8.
// ---------------------------------------------------------------------------
__global__ __launch_bounds__(256) void gemv_wmma_kernel(
    const float* __restrict__ W, const float* __restrict__ x,
    const float* __restrict__ bias, float* __restrict__ y,
    int R, int K, int ksplit)
{
    __shared__ float sx[2048];                 // klen <= 2048
    const int klen  = K / ksplit;
    const int kbase = blockIdx.y * klen;
    for (int i = threadIdx.x; i < klen; i += 256) sx[i] = x[kbase + i];
    __syncthreads();

    const int wave  = threadIdx.x >> 5;        // 0..7 (wave32)
    const int lane  = threadIdx.x & 31;
    const int row0  = (blockIdx.x * 8 + wave) * 16;
    const int m     = lane & 15;               // row within tile (A operand)
    const int khalf = (lane >> 4) << 1;        // lanes 0-15 -> k+{0,1}; 16-31 -> k+{2,3}
    const int rowm  = (row0 + m < R) ? (row0 + m) : (R - 1);   // safe clamp
    const float* wrow = W + (size_t)rowm * (size_t)K + kbase;

    v8f acc0 = {0.f, 0.f, 0.f, 0.f, 0.f, 0.f, 0.f, 0.f};
    v8f acc1 = {0.f, 0.f, 0.f, 0.f, 0.f, 0.f, 0.f, 0.f};

    for (int k = 0; k < klen; k += 8) {
        v2f a0 = *(const v2f*)(wrow + k + khalf);          // global b64, aligned
        v2f b0 = *(const v2f*)(sx   + k + khalf);          // LDS b64
        acc0 = __builtin_amdgcn_wmma_f32_16x16x4_f32(
                   false, a0, false, b0, (short)0, acc0, false, false);
        v2f a1 = *(const v2f*)(wrow + k + 4 + khalf);
        v2f b1 = *(const v2f*)(sx   + k + 4 + khalf);
        acc1 = __builtin_amdgcn_wmma_f32_16x16x4_f32(
                   false, a1, false, b1, (short)0, acc1, false, false);
    }
    v8f acc = acc0 + acc1;                     // D matrices add elementwise

    // D layout: VGPR r holds M=r (lanes 0-15) / M=r+8 (lanes 16-31), N=lane%16.
    // Column N=0 lives in lane 0 (rows 0..7) and lane 16 (rows 8..15).
    if ((lane & 15) == 0) {
        const int mbase = row0 + ((lane >> 4) << 3);
        if (ksplit == 1) {
#pragma unroll
            for (int r = 0; r < 8; ++r) {
                const int row = mbase + r;
                if (row < R) y[row] = acc[r] + bias[row];
            }
        } else {
            float* yp = y + (size_t)blockIdx.y * (size_t)R;
#pragma unroll
            for (int r = 0; r < 8; ++r) {
                const int row = mbase + r;
                if (row < R) yp[row] = acc[r];
            }
        }
    }
}

// Reduce ksplit partials + bias (+ optional relu):  y[r] = act(sum_i part[i*R+r] + bias[r])
template <int ACT>
__global__ void k_gemv_reduce(const float* __restrict__ part, const float* __restrict__ bias,
                              float* __restrict__ y, int R, int S)
{
    const int r = blockIdx.x * blockDim.x + threadIdx.x;
    if (r < R) {
        float s = bias[r];
        for (int i = 0; i < S; ++i) s += part[(size_t)i * R + r];
        if (ACT == 1) s = fmaxf(s, 0.0f);
        y[r] = s;
    }
}

// ---------------------------------------------------------------------------
// cat_eh = [ emb[idx], hidden ]   (2H floats)
// ---------------------------------------------------------------------------
__global__ void k_cat_eh(const float* __restrict__ emb, const int* __restrict__ idx,
                         const float* __restrict__ hidden, float* __restrict__ cat_eh)
{
    const int i = blockIdx.x * blockDim.x + threadIdx.x;
    const int t = idx[0];
    if (i < Hdim)            cat_eh[i] = emb[(size_t)t * Hdim + i];
    else if (i < 2 * Hdim)   cat_eh[i] = hidden[i - Hdim];
}

// ---------------------------------------------------------------------------
// softmax over L=512 logits (in place) + copy weights to d_out slot
// ---------------------------------------------------------------------------
__global__ __launch_bounds__(Ldim) void k_softmax_attn(float* __restrict__ logits,
                                                       float* __restrict__ out_aw)
{
    __shared__ float s[Ldim];
    const int t = threadIdx.x;
    const float v = logits[t];
    s[t] = v; __syncthreads();
    for (int o = Ldim / 2; o > 0; o >>= 1) {
        if (t < o) s[t] = fmaxf(s[t], s[t + o]);
        __syncthreads();
    }
    const float mx = s[0]; __syncthreads();
    const float e = expf(v - mx);
    s[t] = e; __syncthreads();
    for (int o = Ldim / 2; o > 0; o >>= 1) {
        if (t < o) s[t] += s[t + o];
        __syncthreads();
    }
    const float w = e / s[0];
    logits[t] = w;
    out_aw[t] = w;
}

// ---------------------------------------------------------------------------
// attn_applied partials: block (hblk, lc) sums 64 encoder rows into
// part[lc*H + h].  Coalesced: fixed l -> consecutive addresses across threads.
// ---------------------------------------------------------------------------
__global__ void k_attn_partial(const float* __restrict__ aw, const float* __restrict__ enc,
                               float* __restrict__ part)
{
    const int h  = blockIdx.x * blockDim.x + threadIdx.x;   // 0..1023
    const int lc = blockIdx.y;                              // 0..7
    if (h < Hdim) {
        float s = 0.0f;
        const int l0 = lc * (Ldim / 8);
        for (int l = l0; l < l0 + (Ldim / 8); ++l)
            s = fmaf(aw[l], enc[(size_t)l * Hdim + h], s);
        part[(size_t)lc * Hdim + h] = s;
    }
}

__global__ void k_attn_combine(const float* __restrict__ part, const float* __restrict__ cat_eh,
                               float* __restrict__ cat2)
{
    const int h = blockIdx.x * blockDim.x + threadIdx.x;
    if (h < Hdim) {
        float s = 0.0f;
        for (int i = 0; i < 8; ++i) s += part[(size_t)i * Hdim + h];
        cat2[h]        = cat_eh[h];   // embedded half
        cat2[Hdim + h] = s;           // attention context half
    }
}

// ---------------------------------------------------------------------------
// GRU gate math (PyTorch r,z,n ordering); writes h_new to ws and d_out slot
// ---------------------------------------------------------------------------
__global__ void k_gru(const float* __restrict__ gi, const float* __restrict__ gh,
                      const float* __restrict__ hprev,
                      float* __restrict__ hnew, float* __restrict__ out_h)
{
    const int j = blockIdx.x * blockDim.x + threadIdx.x;
    if (j < Hdim) {
        const float r = 1.0f / (1.0f + expf(-(gi[j] + gh[j])));
        const float z = 1.0f / (1.0f + expf(-(gi[Hdim + j] + gh[Hdim + j])));
        const float n = tanhf(gi[2 * Hdim + j] + r * gh[2 * Hdim + j]);
        const float h = hprev[j];
        const float v = (1.0f - z) * n + z * h;
        hnew[j]  = v;
        out_h[j] = v;
    }
}

// ---------------------------------------------------------------------------
// Parallel online logsumexp over V values.
// Merge rule for (m,s) pairs:  m' = max(m1,m2); s' = s1*e^(m1-m') + s2*e^(m2-m')
// ---------------------------------------------------------------------------
__global__ __launch_bounds__(256) void k_lse_partial(const float* __restrict__ vals, int n,
                                                     float* __restrict__ pm, float* __restrict__ ps)
{
    __shared__ float sm[256], ss[256];
    const int t = threadIdx.x;
    const int stride = gridDim.x * 256;
    float m = -3.402823466e38f, s = 0.0f;
    for (int i = blockIdx.x * 256 + t; i < n; i += stride) {
        const float v = vals[i];
        if (v > m) { s = s * expf(m - v) + 1.0f; m = v; }
        else       { s += expf(v - m); }
    }
    sm[t] = m; ss[t] = s; __syncthreads();
    for (int o = 128; o > 0; o >>= 1) {
        if (t < o) {
            const float m1 = sm[t], m2 = sm[t + o];
            const float mm = fmaxf(m1, m2);
            ss[t] = ss[t] * expf(m1 - mm) + ss[t + o] * expf(m2 - mm);
            sm[t] = mm;
        }
        __syncthreads();
    }
    if (t == 0) { pm[blockIdx.x] = sm[0]; ps[blockIdx.x] = ss[0]; }
}

__global__ void k_lse_final(const float* __restrict__ pm, const float* __restrict__ ps,
                            float* __restrict__ lse)
{
    __shared__ float sm[64], ss[64];
    const int t = threadIdx.x;   // 64 threads
    sm[t] = pm[t]; ss[t] = ps[t]; __syncthreads();
    for (int o = 32; o > 0; o >>= 1) {
        if (t < o) {
            const float m1 = sm[t], m2 = sm[t + o];
            const float mm = fmaxf(m1, m2);
            ss[t] = ss[t] * expf(m1 - mm) + ss[t + o] * expf(m2 - mm);
            sm[t] = mm;
        }
        __syncthreads();
    }
    if (t == 0) lse[0] = sm[0] + logf(ss[0]);
}

__global__ void k_lse_apply(float* __restrict__ vals, int n, const float* __restrict__ lse)
{
    const int i = blockIdx.x * blockDim.x + threadIdx.x;
    if (i < n) vals[i] -= lse[0];
}

// ---------------------------------------------------------------------------
extern "C" void kernel_launch(void* const* d_in, const int* in_sizes, int n_in,
                              void* d_out, int out_size, void* d_ws, size_t ws_size,
                              hipStream_t stream)
{
    const int*   idx    = (const int*)  d_in[0];
    const float* hidden = (const float*)d_in[1];   // [H]
    const float* enc    = (const float*)d_in[2];   // [L,H]
    const float* emb    = (const float*)d_in[3];   // [V,H]
    const float* attn_W = (const float*)d_in[4];   // [L,2H]
    const float* attn_b = (const float*)d_in[5];   // [L]
    const float* comb_W = (const float*)d_in[6];   // [H,2H]
    const float* comb_b = (const float*)d_in[7];   // [H]
    const float* W_ih   = (const float*)d_in[8];   // [3H,H]
    const float* W_hh   = (const float*)d_in[9];   // [3H,H]
    const float* b_ih   = (const float*)d_in[10];  // [3H]
    const float* b_hh   = (const float*)d_in[11];  // [3H]
    const float* out_W  = (const float*)d_in[12];  // [V,H]
    const float* out_b  = (const float*)d_in[13];  // [V]

    float* out = (float*)d_out;
    float* out_logits = out;                 // [V]  log_softmax
    float* out_hnew   = out + Vdim;          // [H]  h_new
    float* out_aw     = out + Vdim + Hdim;   // [L]  attn_weights

    float* ws     = (float*)d_ws;
    float* cat_eh = ws;               // 2048
    float* alog   = ws + 2048;        // 512 (logits -> weights, in place)
    float* cat2   = ws + 2560;        // 2048
    float* gi     = ws + 4608;        // 3072
    float* gh     = ws + 7680;        // 3072
    float* xbuf   = ws + 10752;       // 1024 (relu(combine))
    float* hnew   = ws + 11776;       // 1024
    float* part   = ws + 12800;       // 12288 (reused by each split stage)
    float* pm     = ws + 25088;       // 64
    float* ps     = ws + 25152;       // 64
    float* lse    = ws + 25216;       // 1

    // 1. embedded + concat with hidden
    k_cat_eh<<<8, 256, 0, stream>>>(emb, idx, hidden, cat_eh);

    // 2. attention logits (R=512, K=2048, split 8 -> 32 blocks)
    gemv_wmma_kernel<<<dim3(4, 8), 256, 0, stream>>>(attn_W, cat_eh, attn_b, part, Ldim, 2 * Hdim, 8);
    k_gemv_reduce<0><<<2, 256, 0, stream>>>(part, attn_b, alog, Ldim, 8);

    // 3. softmax -> attn_weights (also into d_out)
    k_softmax_attn<<<1, Ldim, 0, stream>>>(alog, out_aw);

    // 4. attn_applied (split over 8 L-chunks) + build [embedded, attn_applied]
    k_attn_partial<<<dim3(4, 8), 256, 0, stream>>>(alog, enc, part);
    k_attn_combine<<<4, 256, 0, stream>>>(part, cat_eh, cat2);

    // 5. combine + relu (R=1024, K=2048, split 8 -> 64 blocks)
    gemv_wmma_kernel<<<dim3(8, 8), 256, 0, stream>>>(comb_W, cat2, comb_b, part, Hdim, 2 * Hdim, 8);
    k_gemv_reduce<1><<<4, 256, 0, stream>>>(part, comb_b, xbuf, Hdim, 8);

    // 6. GRU projections (R=3072, K=1024, split 4 -> 96 blocks each)
    gemv_wmma_kernel<<<dim3(24, 4), 256, 0, stream>>>(W_ih, xbuf, b_ih, part, 3 * Hdim, Hdim, 4);
    k_gemv_reduce<0><<<12, 256, 0, stream>>>(part, b_ih, gi, 3 * Hdim, 4);
    gemv_wmma_kernel<<<dim3(24, 4), 256, 0, stream>>>(W_hh, hidden, b_hh, part, 3 * Hdim, Hdim, 4);
    k_gemv_reduce<0><<<12, 256, 0, stream>>>(part, b_hh, gh, 3 * Hdim, 4);

    // 7. gates -> h_new (also into d_out)
    k_gru<<<4, 256, 0, stream>>>(gi, gh, hidden, hnew, out_hnew);

    // 8. output projection: the 206MB device-wide stream (R=50257, K=1024)
    gemv_wmma_kernel<<<dim3((Vdim + 127) / 128, 1), 256, 0, stream>>>(out_W, hnew, out_b, out_logits, Vdim, Hdim, 1);

    // 9. parallel log_softmax on d_out[0:V]
    k_lse_partial<<<64, 256, 0, stream>>>(out_logits, Vdim, pm, ps);
    k_lse_final<<<1, 64, 0, stream>>>(pm, ps, lse);
    k_lse_apply<<<(Vdim + 255) / 256, 256, 0, stream>>>(out_logits, Vdim, lse);
}